// Additive_Attention_31748398252800
// MI455X (gfx1250) — compile-verified
//
#include <hip/hip_runtime.h>
#include <hip/hip_bf16.h>

typedef __attribute__((ext_vector_type(2))) float v2f;
typedef __attribute__((ext_vector_type(8))) float v8f;

constexpr int B_   = 4;
constexpr int D_   = 96;
constexpr int NSEQ = 512;
constexpr int ATTN = 256;

__device__ __forceinline__ float fast_tanh(float x) {
#if __has_builtin(__builtin_amdgcn_tanhf)
    return __builtin_amdgcn_tanhf(x);
#elif __has_builtin(__builtin_amdgcn_tanh_f32)
    return __builtin_amdgcn_tanh_f32(x);
#else
    return tanhf(x);
#endif
}

// ---------------------------------------------------------------------------
// Kernel 1: g[b,n,a] = sum_d x[b,d,n] * W[a,d]  (+ bg[a] if add_bias)
// One wave per 16x16 output tile, K-loop over D in steps of 4 using
// V_WMMA_F32_16X16X4_F32.
// A fragment (16x4): lanes 0-15 rows 0-15; VGPR0 holds K = 2*(lane/16),
// VGPR1 holds K+1.  B fragment (4x16): mirrored (cols across lanes).
// C/D: acc[v] -> row = v + 8*(lane/16), col = lane%16.
// ---------------------------------------------------------------------------
__global__ __launch_bounds__(128) void gemm_g(const float* __restrict__ x,
                                              const float* __restrict__ W,
                                              const float* __restrict__ bg,
                                              float* __restrict__ g,
                                              int add_bias)
{
    const int tiles_a = ATTN / 16;          // 16
    const int tiles_n = NSEQ / 16;          // 32
    int wave = (blockIdx.x * blockDim.x + threadIdx.x) >> 5;
    int lane = threadIdx.x & 31;
    int b  = wave / (tiles_n * tiles_a);
    int r  = wave % (tiles_n * tiles_a);
    int n0 = (r / tiles_a) * 16;
    int a0 = (r % tiles_a) * 16;

    int half = lane >> 4;                   // 0 or 1
    int l16  = lane & 15;
    int kb   = half * 2;                    // K offset held by this lane

    const float* xb = x + (size_t)b * D_ * NSEQ;

    v8f acc = {};
    for (int k0 = 0; k0 < D_; k0 += 4) {
        int k = k0 + kb;
        v2f a, bf;
        // A[row][k], A[row][k+1] : row = l16, from xt (x transposed)
        a.x = xb[(size_t)(k    ) * NSEQ + (n0 + l16)];
        a.y = xb[(size_t)(k + 1) * NSEQ + (n0 + l16)];
        // B[k][col] = W[a0+col][k] : consecutive k -> float2 load
        const float2 wv = *(const float2*)&W[(size_t)(a0 + l16) * D_ + k];
        bf.x = wv.x;
        bf.y = wv.y;
        acc = __builtin_amdgcn_wmma_f32_16x16x4_f32(
            false, a, false, bf, (short)0, acc, false, false);
    }

    float bias = add_bias ? bg[a0 + l16] : 0.0f;
    float* gp = g + (size_t)b * NSEQ * ATTN;
#pragma unroll
    for (int v = 0; v < 8; ++v) {
        int m = v + half * 8;
        gp[(size_t)(n0 + m) * ATTN + (a0 + l16)] = acc[v] + bias;
    }
}

// ---------------------------------------------------------------------------
// Kernel 2: att[b,n,m] = sigmoid( sum_a Wa_w[a]*tanh(g1[b,n,a]+g2[b,m,a])
//                                 + Wa_b + ba )
// (bg already folded into g1.)  16x16 pair tile per 256-thread block; g1/g2
// rows staged in LDS with pitch 260 words so float4 reads across the 16
// m-lanes touch all 64 banks without conflict.  This stage is the runtime
// hot spot (B*N*N*A = 268M tanh+FMA) and is inherently VALU/trans work.
// ---------------------------------------------------------------------------
#define PITCH 260

__global__ __launch_bounds__(256) void pair_scores(const float* __restrict__ g1,
                                                   const float* __restrict__ g2,
                                                   const float* __restrict__ Wa_w,
                                                   const float* __restrict__ Wa_b,
                                                   const float* __restrict__ ba,
                                                   float* __restrict__ att)
{
    __shared__ float s1[16 * PITCH];
    __shared__ float s2[16 * PITCH];
    __shared__ float sw[ATTN];

    const int tiles = NSEQ / 16;            // 32
    int b  = blockIdx.x / (tiles * tiles);
    int r  = blockIdx.x % (tiles * tiles);
    int n0 = (r / tiles) * 16;
    int m0 = (r % tiles) * 16;

    int t = threadIdx.x;
    sw[t] = Wa_w[t];                        // ATTN == blockDim == 256

    {   // stage 16x256 tiles of g1 and g2 (coalesced float4 loads)
        int row = t >> 4;                   // 0..15
        int l16 = t & 15;
        const float* p1 = g1 + ((size_t)b * NSEQ + n0 + row) * ATTN;
        const float* p2 = g2 + ((size_t)b * NSEQ + m0 + row) * ATTN;
#pragma unroll
        for (int j = 0; j < 4; ++j) {
            int a = (l16 + j * 16) * 4;
            *(float4*)&s1[row * PITCH + a] = *(const float4*)&p1[a];
            *(float4*)&s2[row * PITCH + a] = *(const float4*)&p2[a];
        }
    }
    __syncthreads();

    int tn = t >> 4;                        // row within tile
    int tm = t & 15;                        // col within tile

    float acc = 0.0f;
#pragma unroll 4
    for (int a = 0; a < ATTN; a += 4) {
        float4 w  = *(const float4*)&sw[a];
        float4 v1 = *(const float4*)&s1[tn * PITCH + a];
        float4 v2 = *(const float4*)&s2[tm * PITCH + a];
        acc += w.x * fast_tanh(v1.x + v2.x);
        acc += w.y * fast_tanh(v1.y + v2.y);
        acc += w.z * fast_tanh(v1.z + v2.z);
        acc += w.w * fast_tanh(v1.w + v2.w);
    }

    float s = acc + Wa_b[0] + ba[0];
    float sig = 1.0f / (1.0f + __expf(-s));
    att[((size_t)b * NSEQ + n0 + tn) * NSEQ + (m0 + tm)] = sig;
}

// ---------------------------------------------------------------------------
// Kernel 3: out[b,n,d] = sum_m att[b,n,m] * x[b,d,m]
// One wave per 16x16 tile, K = 512 in steps of 4 (128 WMMA per tile).
// ---------------------------------------------------------------------------
__global__ __launch_bounds__(128) void gemm_out(const float* __restrict__ att,
                                                const float* __restrict__ x,
                                                float* __restrict__ out)
{
    const int tiles_d = D_ / 16;            // 6
    const int tiles_n = NSEQ / 16;          // 32
    int wave = (blockIdx.x * blockDim.x + threadIdx.x) >> 5;
    int lane = threadIdx.x & 31;
    int b  = wave / (tiles_n * tiles_d);
    int r  = wave % (tiles_n * tiles_d);
    int n0 = (r / tiles_d) * 16;
    int d0 = (r % tiles_d) * 16;

    int half = lane >> 4;
    int l16  = lane & 15;
    int kb   = half * 2;

    const float* attb = att + (size_t)b * NSEQ * NSEQ;
    const float* xb   = x   + (size_t)b * D_ * NSEQ;

    v8f acc = {};
    for (int k0 = 0; k0 < NSEQ; k0 += 4) {
        int k = k0 + kb;
        v2f a, bf;
        const float2 av = *(const float2*)&attb[(size_t)(n0 + l16) * NSEQ + k];
        const float2 bv = *(const float2*)&xb[(size_t)(d0 + l16) * NSEQ + k];
        a.x = av.x;  a.y = av.y;            // A[row][k], A[row][k+1]
        bf.x = bv.x; bf.y = bv.y;           // B[k][col] = xt[k][d0+col]
        acc = __builtin_amdgcn_wmma_f32_16x16x4_f32(
            false, a, false, bf, (short)0, acc, false, false);
    }

    float* ob = out + (size_t)b * NSEQ * D_;
#pragma unroll
    for (int v = 0; v < 8; ++v) {
        int m = v + half * 8;
        ob[(size_t)(n0 + m) * D_ + (d0 + l16)] = acc[v];
    }
}

// ---------------------------------------------------------------------------
extern "C" void kernel_launch(void* const* d_in, const int* in_sizes, int n_in,
                              void* d_out, int out_size, void* d_ws, size_t ws_size,
                              hipStream_t stream) {
    const float* x    = (const float*)d_in[0];   // (B, D, N)
    const float* Wg1  = (const float*)d_in[1];   // (ATTN, D)
    const float* Wg2  = (const float*)d_in[2];   // (ATTN, D)
    const float* bg   = (const float*)d_in[3];   // (ATTN,)
    const float* Wa_w = (const float*)d_in[4];   // (1, ATTN)
    const float* Wa_b = (const float*)d_in[5];   // (1,)
    const float* ba   = (const float*)d_in[6];   // (1,)
    float* out = (float*)d_out;                  // (B, N, D)

    float* ws  = (float*)d_ws;
    float* g1  = ws;                                       // B*N*ATTN
    float* g2  = g1 + (size_t)B_ * NSEQ * ATTN;            // B*N*ATTN
    float* att = g2 + (size_t)B_ * NSEQ * ATTN;            // B*N*N

    // Kernel 1: two projection GEMMs (bg folded into g1)
    {
        int waves  = B_ * (NSEQ / 16) * (ATTN / 16);       // 2048
        int blocks = waves / 4;                            // 4 waves / block
        gemm_g<<<blocks, 128, 0, stream>>>(x, Wg1, bg, g1, 1);
        gemm_g<<<blocks, 128, 0, stream>>>(x, Wg2, bg, g2, 0);
    }

    // Kernel 2: pairwise additive-attention scores (dominant cost)
    {
        int blocks = B_ * (NSEQ / 16) * (NSEQ / 16);       // 4096
        pair_scores<<<blocks, 256, 0, stream>>>(g1, g2, Wa_w, Wa_b, ba, att);
    }

    // Kernel 3: out = att @ xt
    {
        int waves  = B_ * (NSEQ / 16) * (D_ / 16);         // 768
        int blocks = waves / 4;                            // 192
        gemm_out<<<blocks, 128, 0, stream>>>(att, x, out);
    }
}